// G2_GNN_91190745629151
// MI455X (gfx1250) — compile-verified
//
#include <hip/hip_runtime.h>
#include <math.h>

typedef float v2f __attribute__((ext_vector_type(2)));
typedef float v8f __attribute__((ext_vector_type(8)));

#define NHID 64

// ---------------------------------------------------------------------------
// WMMA fp32 GEMM:  C[M,64] = A[M,K] @ W[K,64]  (+bias, optional ReLU)
// One wave32 computes a 16x64 tile using V_WMMA_F32_16X16X4_F32.
// Fragment layouts per CDNA5 ISA 7.12.2:
//   A 16x4 f32 : lanes 0-15 -> M=row0+lane (K pair 0,1); lanes 16-31 K pair 2,3
//   B 4x16 f32 : lane%16 -> N; lane/16 selects K pair {0,1} or {2,3}
//   C 16x16 f32: lane%16 -> N; VGPR i -> M = i + (lane/16)*8
// ---------------------------------------------------------------------------
__global__ __launch_bounds__(128) void g2_gemm_n64(
    const float* __restrict__ A, const float* __restrict__ W,
    const float* __restrict__ bias, float* __restrict__ C,
    int M, int K, int do_relu)
{
    int gtid  = blockIdx.x * blockDim.x + threadIdx.x;
    int wave  = gtid >> 5;            // wave-uniform
    int lane  = threadIdx.x & 31;
    int row0  = wave * 16;
    if (row0 >= M) return;            // wave-uniform exit: EXEC all-ones below

    int lmod  = lane & 15;
    int khalf = (lane >> 4) * 2;      // 0 or 2
    int r     = row0 + lmod;

    v8f acc[4] = {};
    for (int k = 0; k < K; k += 4) {
        int kk = k + khalf;
        v2f a = *(const v2f*)(A + (size_t)r * K + kk);
#pragma unroll
        for (int t = 0; t < 4; ++t) {
            int n = t * 16 + lmod;
            v2f b;
            b.x = W[(size_t)kk * NHID + n];
            b.y = W[(size_t)(kk + 1) * NHID + n];
            acc[t] = __builtin_amdgcn_wmma_f32_16x16x4_f32(
                false, a, false, b, (short)0, acc[t], false, false);
        }
    }

    int mbase = row0 + (lane >> 4) * 8;
#pragma unroll
    for (int t = 0; t < 4; ++t) {
        int n = t * 16 + lmod;
        float bv = bias ? bias[n] : 0.0f;
#pragma unroll
        for (int i = 0; i < 8; ++i) {
            float v = acc[t][i] + bv;
            if (do_relu) v = fmaxf(v, 0.0f);
            C[(size_t)(mbase + i) * NHID + n] = v;
        }
    }
}

// Fused dual GEMM (shared A): C1 = A@W1, C2 = A@W2  (K = 64, no bias/relu;
// bias+relu are applied later in the aggregation stage). Loads each A
// fragment once, feeds two WMMA chains.
__global__ __launch_bounds__(128) void g2_gemm_n64_dual(
    const float* __restrict__ A,
    const float* __restrict__ W1, const float* __restrict__ W2,
    float* __restrict__ C1, float* __restrict__ C2, int M)
{
    const int K = NHID;
    int gtid  = blockIdx.x * blockDim.x + threadIdx.x;
    int wave  = gtid >> 5;
    int lane  = threadIdx.x & 31;
    int row0  = wave * 16;
    if (row0 >= M) return;            // wave-uniform

    int lmod  = lane & 15;
    int khalf = (lane >> 4) * 2;
    int r     = row0 + lmod;

    v8f acc1[4] = {};
    v8f acc2[4] = {};
    for (int k = 0; k < K; k += 4) {
        int kk = k + khalf;
        v2f a = *(const v2f*)(A + (size_t)r * K + kk);
#pragma unroll
        for (int t = 0; t < 4; ++t) {
            int n = t * 16 + lmod;
            v2f b1, b2;
            b1.x = W1[(size_t)kk * NHID + n];
            b1.y = W1[(size_t)(kk + 1) * NHID + n];
            b2.x = W2[(size_t)kk * NHID + n];
            b2.y = W2[(size_t)(kk + 1) * NHID + n];
            acc1[t] = __builtin_amdgcn_wmma_f32_16x16x4_f32(
                false, a, false, b1, (short)0, acc1[t], false, false);
            acc2[t] = __builtin_amdgcn_wmma_f32_16x16x4_f32(
                false, a, false, b2, (short)0, acc2[t], false, false);
        }
    }

    int mbase = row0 + (lane >> 4) * 8;
#pragma unroll
    for (int t = 0; t < 4; ++t) {
        int n = t * 16 + lmod;
#pragma unroll
        for (int i = 0; i < 8; ++i) {
            C1[(size_t)(mbase + i) * NHID + n] = acc1[t][i];
            C2[(size_t)(mbase + i) * NHID + n] = acc2[t][i];
        }
    }
}

// ---------------------------------------------------------------------------
// Graph preprocessing (once per launch)
// ---------------------------------------------------------------------------
__global__ void g2_histogram(const int* __restrict__ src, const int* __restrict__ dst,
                             int E, int* __restrict__ indeg, int* __restrict__ outdeg)
{
    int e = blockIdx.x * blockDim.x + threadIdx.x;
    if (e >= E) return;
    atomicAdd(&indeg[dst[e]], 1);
    atomicAdd(&outdeg[src[e]], 1);
}

__global__ void g2_norms(const int* __restrict__ indeg, const int* __restrict__ outdeg,
                         float* __restrict__ dis, float* __restrict__ cntf, int n)
{
    int i = blockIdx.x * blockDim.x + threadIdx.x;
    if (i >= n) return;
    dis[i]  = rsqrtf((float)indeg[i] + 1.0f);      // +1 self loop; deg>0 always
    cntf[i] = fmaxf((float)outdeg[i], 1.0f);
}

// Single-block exclusive scan (n ~ 50k: 49 chunk iterations; runs once/launch)
__global__ __launch_bounds__(1024) void g2_scan(const int* __restrict__ cnt,
                                                int* __restrict__ off, int n)
{
    __shared__ int smem[1024];
    __shared__ int running;
    if (threadIdx.x == 0) running = 0;
    __syncthreads();
    for (int base = 0; base < n; base += 1024) {
        int i = base + threadIdx.x;
        int v = (i < n) ? cnt[i] : 0;
        smem[threadIdx.x] = v;
        __syncthreads();
        for (int d = 1; d < 1024; d <<= 1) {
            int t = (threadIdx.x >= d) ? smem[threadIdx.x - d] : 0;
            __syncthreads();
            smem[threadIdx.x] += t;
            __syncthreads();
        }
        int ex = running + smem[threadIdx.x] - v;   // read running
        __syncthreads();
        if (i < n) off[i] = ex;
        if (threadIdx.x == 1023) running += smem[1023];
        __syncthreads();
    }
    if (threadIdx.x == 0) off[n] = running;
}

__global__ void g2_csr_fill(const int* __restrict__ src, const int* __restrict__ dst, int E,
                            int* __restrict__ cur_d, int* __restrict__ col_d,
                            int* __restrict__ cur_s, int* __restrict__ col_s)
{
    int e = blockIdx.x * blockDim.x + threadIdx.x;
    if (e >= E) return;
    int s = src[e], d = dst[e];
    col_d[atomicAdd(&cur_d[d], 1)] = s;   // in-edges of d  -> gather sources
    col_s[atomicAdd(&cur_s[s], 1)] = d;   // out-edges of s -> gate neighbors
}

// ---------------------------------------------------------------------------
// GCNConv aggregation: out[d] = relu( dis[d]*( dis[d]*XW[d] + sum_s dis[s]*XW[s] ) + b )
// thread = (node, float4 chunk): each neighbor gather is one b128 load per
// lane (512B per wave) instead of four b32s; everything is L2-resident.
// ---------------------------------------------------------------------------
__global__ __launch_bounds__(256) void g2_aggregate(
    const float4* __restrict__ XW4, const float4* __restrict__ bias4,
    const float* __restrict__ dis, const int* __restrict__ off,
    const int* __restrict__ col, float4* __restrict__ out4, int n)
{
    int t = blockIdx.x * blockDim.x + threadIdx.x;
    int node = t >> 4;          // 16 chunks of float4 per 64-feature row
    int c    = t & 15;
    if (node >= n) return;
    float dd = dis[node];
    float4 v = XW4[(size_t)node * 16 + c];
    float4 acc = make_float4(dd * v.x, dd * v.y, dd * v.z, dd * v.w); // self loop
    int e1 = off[node + 1];
    for (int e = off[node]; e < e1; ++e) {
        int s   = col[e];
        float w = dis[s];
        float4 u = XW4[(size_t)s * 16 + c];
        acc.x += w * u.x;  acc.y += w * u.y;
        acc.z += w * u.z;  acc.w += w * u.w;
    }
    float4 b = bias4[c];
    float4 o;
    o.x = fmaxf(dd * acc.x + b.x, 0.0f);
    o.y = fmaxf(dd * acc.y + b.y, 0.0f);
    o.z = fmaxf(dd * acc.z + b.z, 0.0f);
    o.w = fmaxf(dd * acc.w + b.w, 0.0f);
    out4[(size_t)node * 16 + c] = o;
}

// Gate: tau = tanh( sum_{d in out(s)} (xg[s]-xg[d])^2 / max(cnt,1) );
//       X = (1-tau)*X + tau*Xc   (in place)
__global__ __launch_bounds__(256) void g2_gate_update(
    const float4* __restrict__ xg4, const float4* __restrict__ Xc4,
    float4* __restrict__ X4, const float* __restrict__ cntf,
    const int* __restrict__ off_s, const int* __restrict__ col_s, int n)
{
    int t = blockIdx.x * blockDim.x + threadIdx.x;
    int node = t >> 4;
    int c    = t & 15;
    if (node >= n) return;
    float4 xs  = xg4[(size_t)node * 16 + c];
    float4 acc = make_float4(0.f, 0.f, 0.f, 0.f);
    int e1 = off_s[node + 1];
    for (int e = off_s[node]; e < e1; ++e) {
        float4 xd = xg4[(size_t)col_s[e] * 16 + c];
        float dx = xs.x - xd.x, dy = xs.y - xd.y;
        float dz = xs.z - xd.z, dw = xs.w - xd.w;
        acc.x += dx * dx;  acc.y += dy * dy;
        acc.z += dz * dz;  acc.w += dw * dw;     // |.|^P, P=2
    }
    float rc = 1.0f / cntf[node];
    float4 tau;
    tau.x = tanhf(acc.x * rc);  tau.y = tanhf(acc.y * rc);
    tau.z = tanhf(acc.z * rc);  tau.w = tanhf(acc.w * rc);
    size_t idx = (size_t)node * 16 + c;
    float4 xv = X4[idx];
    float4 xc = Xc4[idx];
    float4 o;
    o.x = (1.0f - tau.x) * xv.x + tau.x * xc.x;
    o.y = (1.0f - tau.y) * xv.y + tau.y * xc.y;
    o.z = (1.0f - tau.z) * xv.z + tau.z * xc.z;
    o.w = (1.0f - tau.w) * xv.w + tau.w * xc.w;
    X4[idx] = o;
}

// ---------------------------------------------------------------------------
extern "C" void kernel_launch(void* const* d_in, const int* in_sizes, int n_in,
                              void* d_out, int out_size, void* d_ws, size_t ws_size,
                              hipStream_t stream)
{
    const float* x      = (const float*)d_in[0];
    const int*   eidx   = (const int*)  d_in[1];
    const float* enc_w  = (const float*)d_in[2];
    const float* enc_b  = (const float*)d_in[3];
    const float* conv_w = (const float*)d_in[4];
    const float* conv_b = (const float*)d_in[5];
    const float* gg_w   = (const float*)d_in[6];
    const float* gg_b   = (const float*)d_in[7];
    const float* dec_w  = (const float*)d_in[8];
    const float* dec_b  = (const float*)d_in[9];

    const int n     = in_sizes[0] / 256;     // N_NODES
    const int E     = in_sizes[1] / 2;       // N_EDGES
    const int INDIM = 256;
    const int* src = eidx;
    const int* dst = eidx + E;

    // --- workspace partition (256B aligned) ---
    char* p = (char*)d_ws;
    auto alloc = [&](size_t bytes) -> void* {
        void* r = (void*)p;
        p += (bytes + 255) & ~(size_t)255;
        return r;
    };
    size_t nf = (size_t)n * NHID * sizeof(float);
    float* X     = (float*)alloc(nf);
    float* XWc   = (float*)alloc(nf);
    float* XWg   = (float*)alloc(nf);
    float* Xc    = (float*)alloc(nf);
    float* xg    = (float*)alloc(nf);
    int*   indeg = (int*)  alloc((size_t)n * 4);
    int*   outdg = (int*)  alloc((size_t)n * 4);
    float* dis   = (float*)alloc((size_t)n * 4);
    float* cntf  = (float*)alloc((size_t)n * 4);
    int*   off_d = (int*)  alloc((size_t)(n + 1) * 4);
    int*   off_s = (int*)  alloc((size_t)(n + 1) * 4);
    int*   cur_d = (int*)  alloc((size_t)n * 4);
    int*   cur_s = (int*)  alloc((size_t)n * 4);
    int*   col_d = (int*)  alloc((size_t)E * 4);
    int*   col_s = (int*)  alloc((size_t)E * 4);

    const int TB = 256;
    int ne_blk  = (E + TB - 1) / TB;
    int nn_blk  = (n + TB - 1) / TB;
    int agg_blk = ((size_t)n * 16 + TB - 1) / TB;   // thread per (node, float4)

    // --- graph preprocessing (per launch; deterministic) ---
    hipMemsetAsync(indeg, 0, (size_t)n * 4, stream);
    hipMemsetAsync(outdg, 0, (size_t)n * 4, stream);
    g2_histogram<<<ne_blk, TB, 0, stream>>>(src, dst, E, indeg, outdg);
    g2_norms<<<nn_blk, TB, 0, stream>>>(indeg, outdg, dis, cntf, n);
    g2_scan<<<1, 1024, 0, stream>>>(indeg, off_d, n);
    g2_scan<<<1, 1024, 0, stream>>>(outdg, off_s, n);
    hipMemcpyAsync(cur_d, off_d, (size_t)n * 4, hipMemcpyDeviceToDevice, stream);
    hipMemcpyAsync(cur_s, off_s, (size_t)n * 4, hipMemcpyDeviceToDevice, stream);
    g2_csr_fill<<<ne_blk, TB, 0, stream>>>(src, dst, E, cur_d, col_d, cur_s, col_s);

    // --- encoder: X = relu(x @ enc_w + enc_b) ---
    int waves   = (n + 15) / 16;
    int gemm_bl = (waves * 32 + 127) / 128;
    g2_gemm_n64<<<gemm_bl, 128, 0, stream>>>(x, enc_w, enc_b, X, n, INDIM, 1);

    // --- 4 G2 layers ---
    for (int layer = 0; layer < 4; ++layer) {
        g2_gemm_n64_dual<<<gemm_bl, 128, 0, stream>>>(X, conv_w, gg_w, XWc, XWg, n);
        g2_aggregate<<<agg_blk, TB, 0, stream>>>((const float4*)XWc, (const float4*)conv_b,
                                                 dis, off_d, col_d, (float4*)Xc, n);
        g2_aggregate<<<agg_blk, TB, 0, stream>>>((const float4*)XWg, (const float4*)gg_b,
                                                 dis, off_d, col_d, (float4*)xg, n);
        g2_gate_update<<<agg_blk, TB, 0, stream>>>((const float4*)xg, (const float4*)Xc,
                                                   (float4*)X, cntf, off_s, col_s, n);
    }

    // --- decoder: out = relu(X @ dec_w + dec_b) ---
    g2_gemm_n64<<<gemm_bl, 128, 0, stream>>>(X, dec_w, dec_b, (float*)d_out, n, NHID, 1);
}